// LSTM_Op_27994596836216
// MI455X (gfx1250) — compile-verified
//
#include <hip/hip_runtime.h>
#include <hip/hip_bf16.h>

#define BATCH 32
#define TSTEPS 128
#define CH 64
#define DDIN 32
#define DDOUT 32

typedef float v2f __attribute__((ext_vector_type(2)));
typedef float v8f __attribute__((ext_vector_type(8)));

__device__ __forceinline__ float sigmoidf(float v) {
    return 1.0f / (1.0f + __expf(-v));
}

// LDS-only workgroup barrier: waits for this wave's outstanding LDS ops, then
// signals/waits the workgroup barrier. Unlike __syncthreads() it does NOT
// force LOADcnt->0, so the software-pipelined global weight loads for t+1 can
// remain in flight across the barrier.
__device__ __forceinline__ void lds_barrier() {
    asm volatile(
        "s_wait_dscnt 0x0\n\t"
        "s_barrier_signal -1\n\t"
        "s_barrier_wait -1"
        ::: "memory");
}

// One workgroup per channel c. 8 waves: wave = (gate g in [0,4)) x (row-tile m in [0,2)).
// Activations live in LDS in K-pair-packed layout: P[k>>1][n][k&1], so every
// WMMA B-fragment (f32 16x16x4) is one contiguous 8-byte LDS read.
// Weights AND the x tile are software-pipelined one timestep ahead in registers.
__global__ __launch_bounds__(256) void lstm_grouped_wmma(
    const float* __restrict__ x,      // [B, T, C, DIN]
    const float* __restrict__ xOps,   // [4, T, C, DOUT, DIN]
    const float* __restrict__ hOps,   // [4, T, C, DOUT, DOUT]
    float* __restrict__ out)          // [B, C, DOUT]
{
    const int c    = blockIdx.x;
    const int tid  = threadIdx.x;
    const int lane = tid & 31;
    const int wave = tid >> 5;     // 0..7
    const int g     = wave >> 1;   // gate
    const int mtile = wave & 1;    // 16-row tile of DOUT

    // Packed activation buffers: index = (k>>1)*64 + n*2 + (k&1), 1024 floats each.
    __shared__ float xP[2][DDIN * BATCH];
    __shared__ float hP[2][DDOUT * BATCH];
    __shared__ float gates[4 * DDOUT * BATCH];     // [g][i][b] linear

    for (int e = tid; e < DDOUT * BATCH; e += 256) hP[0][e] = 0.0f;

    // Cell state: thread owns i-pairs (kpair = pp>>5, b = pp&31) for pp = tid, tid+256.
    float cst[4] = {0.0f, 0.0f, 0.0f, 0.0f};

    // f32 WMMA A-fragment addressing: lanes 0-15 hold K {0,1}, lanes 16-31 hold K {2,3}.
    const int m     = (mtile << 4) + (lane & 15);
    const int khalf = lane >> 4;
    const int dn    = lane & 15;
    const int dhalf = lane >> 4;
    const int n2    = (lane & 15) << 1;   // packed column offset

    const float* xw = xOps + ((size_t)g * TSTEPS * CH + c) * (DDOUT * DDIN);
    const float* hw = hOps + ((size_t)g * TSTEPS * CH + c) * (DDOUT * DDOUT);
    const size_t wstride = (size_t)CH * DDOUT * DDIN;   // per-timestep weight stride

    // Per-thread x staging addresses (b = tid>>3, j0 = (tid&7)*4)
    const int xb  = tid >> 3;
    const int xj0 = (tid & 7) << 2;
    const float* xbase = x + (((size_t)xb * TSTEPS) * CH + c) * DDIN + xj0;
    const size_t xtstride = (size_t)CH * DDIN;          // per-timestep x stride

    auto load_w = [&](int t, v2f (&ax)[8], v2f (&ah)[8]) {
        const float* xwt = xw + (size_t)t * wstride;
        const float* hwt = hw + (size_t)t * wstride;
        #pragma unroll
        for (int kk = 0; kk < 8; ++kk) {
            const int col = (kk << 2) + (khalf << 1);
            ax[kk] = *reinterpret_cast<const v2f*>(xwt + m * DDIN + col);
            ah[kk] = *reinterpret_cast<const v2f*>(hwt + m * DDOUT + col);
        }
    };

    v2f axA[8], ahA[8], axB[8], ahB[8];
    float4 xvA, xvB;
    load_w(0, axA, ahA);
    xvA = *reinterpret_cast<const float4*>(xbase);      // x tile for t=0

    // One LSTM timestep. pb selects the activation buffer this step READS;
    // the elementwise phase writes h into buffer pb^1 for the next step.
    auto step = [&](int t, int pb,
                    const float4& xvu, float4& xvl,
                    v2f (&axu)[8], v2f (&ahu)[8],
                    v2f (&axl)[8], v2f (&ahl)[8]) {
        // ---- stage x_t^T (packed) from the register loaded last step ----
        *reinterpret_cast<v2f*>(&xP[pb][((xj0 >> 1) + 0) * (BATCH * 2) + (xb << 1)]) = v2f{xvu.x, xvu.y};
        *reinterpret_cast<v2f*>(&xP[pb][((xj0 >> 1) + 1) * (BATCH * 2) + (xb << 1)]) = v2f{xvu.z, xvu.w};

        // ---- software pipeline: fetch x tile and weights for t+1 ----
        if (t + 1 < TSTEPS) {
            xvl = *reinterpret_cast<const float4*>(xbase + (size_t)(t + 1) * xtstride);
            load_w(t + 1, axl, ahl);
        }
        if (t + 2 < TSTEPS) {
            __builtin_prefetch(xw + (size_t)(t + 2) * wstride + m * DDIN, 0, 2);
            __builtin_prefetch(hw + (size_t)(t + 2) * wstride + m * DDOUT, 0, 2);
        }

        v8f acc0 = {0.f,0.f,0.f,0.f,0.f,0.f,0.f,0.f};
        v8f acc1 = {0.f,0.f,0.f,0.f,0.f,0.f,0.f,0.f};

        lds_barrier();   // xP[pb] and hP[pb] ready; weight loads stay in flight

        // ---- D[i,b] = xop @ x^T + hop @ h^T : 32 wmma ----
        #pragma unroll
        for (int kk = 0; kk < 8; ++kk) {
            const int kp = ((kk << 1) + khalf) * (BATCH * 2);  // packed k-pair row
            const v2f bx0 = *reinterpret_cast<const v2f*>(&xP[pb][kp + n2]);
            const v2f bx1 = *reinterpret_cast<const v2f*>(&xP[pb][kp + 32 + n2]);
            const v2f bh0 = *reinterpret_cast<const v2f*>(&hP[pb][kp + n2]);
            const v2f bh1 = *reinterpret_cast<const v2f*>(&hP[pb][kp + 32 + n2]);

            acc0 = __builtin_amdgcn_wmma_f32_16x16x4_f32(
                       false, axu[kk], false, bx0, (short)0, acc0, false, false);
            acc1 = __builtin_amdgcn_wmma_f32_16x16x4_f32(
                       false, axu[kk], false, bx1, (short)0, acc1, false, false);
            acc0 = __builtin_amdgcn_wmma_f32_16x16x4_f32(
                       false, ahu[kk], false, bh0, (short)0, acc0, false, false);
            acc1 = __builtin_amdgcn_wmma_f32_16x16x4_f32(
                       false, ahu[kk], false, bh1, (short)0, acc1, false, false);
        }

        // ---- scatter gate pre-activations: VGPR r -> M = r + 8*dhalf (+16*mtile) ----
        #pragma unroll
        for (int r = 0; r < 8; ++r) {
            const int mi = (mtile << 4) + r + (dhalf << 3);
            gates[g * (DDOUT * BATCH) + mi * BATCH + dn]      = acc0[r];
            gates[g * (DDOUT * BATCH) + mi * BATCH + 16 + dn] = acc1[r];
        }

        lds_barrier();   // gates visible

        // ---- elementwise cell update; h written packed into buffer pb^1 ----
        #pragma unroll
        for (int q2 = 0; q2 < 2; ++q2) {
            const int pp    = tid + (q2 << 8);   // i-pair index * 32 + b
            const int kpair = pp >> 5;
            const int b     = pp & 31;
            v2f hn;
            #pragma unroll
            for (int di = 0; di < 2; ++di) {
                const int e  = ((kpair << 1) + di) * BATCH + b;
                const float Xi = gates[0 * (DDOUT * BATCH) + e];
                const float Xf = gates[1 * (DDOUT * BATCH) + e];
                const float Xg = gates[2 * (DDOUT * BATCH) + e];
                const float Xo = gates[3 * (DDOUT * BATCH) + e];
                const float ig = sigmoidf(Xi);
                const float fg = sigmoidf(Xf);
                const float gg = tanhf(Xg);
                const float og = sigmoidf(Xo);
                const float cn = fg * cst[(q2 << 1) + di] + ig * gg;
                cst[(q2 << 1) + di] = cn;
                hn[di] = og * tanhf(cn);
            }
            *reinterpret_cast<v2f*>(&hP[pb ^ 1][kpair * (BATCH * 2) + (b << 1)]) = hn;
        }
        // no third barrier: next step's first barrier fences hP[pb^1]/gates reuse
    };

    lds_barrier();   // h0 zero-init visible

    #pragma unroll 1
    for (int t = 0; t < TSTEPS; t += 2) {
        step(t,     0, xvA, xvB, axA, ahA, axB, ahB);
        step(t + 1, 1, xvB, xvA, axB, ahB, axA, ahA);
    }

    // ---- write h_final = hP[0] (written by step t=127): out[b, c, i] ----
    #pragma unroll
    for (int q2 = 0; q2 < 2; ++q2) {
        const int pp    = tid + (q2 << 8);
        const int kpair = pp >> 5;
        const int b     = pp & 31;
        const v2f hv = *reinterpret_cast<const v2f*>(&hP[0][kpair * (BATCH * 2) + (b << 1)]);
        *reinterpret_cast<v2f*>(out + ((size_t)b * CH + c) * DDOUT + (kpair << 1)) = hv;
    }
}

extern "C" void kernel_launch(void* const* d_in, const int* in_sizes, int n_in,
                              void* d_out, int out_size, void* d_ws, size_t ws_size,
                              hipStream_t stream) {
    (void)in_sizes; (void)n_in; (void)out_size; (void)d_ws; (void)ws_size;
    const float* x    = (const float*)d_in[0];
    const float* xOps = (const float*)d_in[1];
    const float* hOps = (const float*)d_in[2];
    float* out = (float*)d_out;
    lstm_grouped_wmma<<<dim3(CH), dim3(256), 0, stream>>>(x, xOps, hOps, out);
}